// DLPCNNLoss_59545426592405
// MI455X (gfx1250) — compile-verified
//
#include <hip/hip_runtime.h>
#include <hip/hip_bf16.h>

// DLPCNN loss: lambda/2 * sum(top-20 smallest same-class pairwise sq-distances)
//            + mean cross-entropy.
// Gram matrix via V_WMMA_F32_16X16X4_F32; A-block resident in 320KB CDNA5 LDS.

#define BATCH   8192
#define DIM     2000
#define NCLS    7
#define TOPK    20
#define BIGV    1e30f
#define WAVES   8
#define ASTRIDE 2002   // padded LDS row stride (dwords): conflict-free b64 reads

typedef __attribute__((ext_vector_type(2))) float v2f;
typedef __attribute__((ext_vector_type(8))) float v8f;

// ---------------------------------------------------------------- row norms
__global__ __launch_bounds__(256) void norms_kernel(const float* __restrict__ xfeat,
                                                    float* __restrict__ sq) {
    __shared__ float red[256];
    const int r = blockIdx.x;
    const float* row = xfeat + (size_t)r * DIM;
    float s = 0.0f;
    for (int k = threadIdx.x; k < DIM; k += 256) {
        float x = row[k];
        s += x * x;
    }
    red[threadIdx.x] = s;
    __syncthreads();
    for (int off = 128; off > 0; off >>= 1) {
        if (threadIdx.x < off) red[threadIdx.x] += red[threadIdx.x + off];
        __syncthreads();
    }
    if (threadIdx.x == 0) sq[r] = red[0];
}

// ---------------------------------------------------------------- cross entropy
__global__ __launch_bounds__(256) void ce_kernel(const float* __restrict__ xsoft,
                                                 const int* __restrict__ y,
                                                 float* __restrict__ ce_part) {
    __shared__ float red[256];
    const int i = blockIdx.x * 256 + threadIdx.x;
    float v = 0.0f;
    if (i < BATCH) {
        const float* row = xsoft + (size_t)i * NCLS;
        float m = row[0];
        #pragma unroll
        for (int c = 1; c < NCLS; ++c) m = fmaxf(m, row[c]);
        float se = 0.0f;
        #pragma unroll
        for (int c = 0; c < NCLS; ++c) se += expf(row[c] - m);
        v = m + logf(se) - row[y[i]];
    }
    red[threadIdx.x] = v;
    __syncthreads();
    for (int off = 128; off > 0; off >>= 1) {
        if (threadIdx.x < off) red[threadIdx.x] += red[threadIdx.x + off];
        __syncthreads();
    }
    if (threadIdx.x == 0) ce_part[blockIdx.x] = red[0];
}

// ---------------------------------------------------------------- lp (WMMA) kernel
// One workgroup = 16 rows. 8 waves split the 512 column tiles (64 each).
__global__ __launch_bounds__(256) void lp_kernel(const float* __restrict__ xfeat,
                                                 const int* __restrict__ y,
                                                 const float* __restrict__ sq,
                                                 float* __restrict__ lp_part) {
    __shared__ float As[16 * ASTRIDE];           // ~125 KB: 16 full feature rows
    __shared__ float tile[WAVES][16][16];        // per-wave distance tile
    __shared__ float heap[WAVES][16][TOPK];      // per-wave per-row top-20 (min set)
    __shared__ int   yrow[16];
    __shared__ float sqrow[16];
    __shared__ float red[16];

    const int tid  = threadIdx.x;
    const int wave = tid >> 5;
    const int lane = tid & 31;
    const int half = lane >> 4;        // 0: K 0..1 | 1: K 2..3  (fp32 A/B frag layout)
    const int n    = lane & 15;        // column within tile / A row within tile
    const int rowbase = blockIdx.x * 16;

    // Stage A block (16 rows x 2000) into LDS, coalesced.
    for (int idx = tid; idx < 16 * DIM; idx += 256) {
        const int r = idx / DIM;
        const int k = idx - r * DIM;
        As[r * ASTRIDE + k] = xfeat[(size_t)(rowbase + r) * DIM + k];
    }
    if (tid < 16) {
        yrow[tid]  = y[rowbase + tid];
        sqrow[tid] = sq[rowbase + tid];
    }
    for (int idx = tid; idx < WAVES * 16 * TOPK; idx += 256)
        (&heap[0][0][0])[idx] = BIGV;
    __syncthreads();

    // Hoist per-lane row metadata (loop-invariant): kills divergent LDS loads
    // in the per-tile epilogue — it becomes straight-line cmp/cndmask/store.
    int   yr[8];
    float sr[8];
    int   gi[8];
    #pragma unroll
    for (int v = 0; v < 8; ++v) {
        const int il = v + 8 * half;             // C/D layout: row index
        yr[v] = yrow[il];
        sr[v] = sqrow[il];
        gi[v] = rowbase + il;
    }

    float maxVal = BIGV;   // live only in lanes 0..15: current max of this row's heap
    int   maxIdx = 0;

    const int tilesPerWave = (BATCH / 16) / WAVES;   // 64
    const float* arow = &As[n * ASTRIDE];
    const int koff = half * 2;

    for (int t = 0; t < tilesPerWave; ++t) {
        const int colbase = (wave + t * WAVES) * 16;
        const int j = colbase + n;
        const float* brow = xfeat + (size_t)j * DIM;

        // Warm L2/L0 for the next tile's B rows.
        if (t + 1 < tilesPerWave) {
            const int jn = (wave + (t + 1) * WAVES) * 16 + n;
            __builtin_prefetch(xfeat + (size_t)jn * DIM, 0, 0);
        }

        v8f acc = {};
        #pragma unroll 4
        for (int k = 0; k < DIM; k += 4) {
            v2f a = *(const v2f*)(arow + k + koff);   // ds_load_2addr_b64 (8B aligned)
            v2f b = *(const v2f*)(brow + k + koff);   // global_load_b64 (L2 resident)
            acc = __builtin_amdgcn_wmma_f32_16x16x4_f32(
                false, a, false, b, (short)0, acc, false, false);
        }

        const int   yj  = y[j];
        const float sqj = sq[j];
        #pragma unroll
        for (int v = 0; v < 8; ++v) {
            float dv = sr[v] + sqj - 2.0f * acc[v];
            const bool valid = (yr[v] == yj) && (gi[v] != j);
            tile[wave][v + 8 * half][n] = valid ? dv : BIGV;
        }
        __syncthreads();

        // One lane per row: fold 16 candidates into this wave's top-20 set.
        if (lane < 16) {
            float mv = maxVal;
            int   mi = maxIdx;
            #pragma unroll 1
            for (int c = 0; c < 16; ++c) {
                const float val = tile[wave][lane][c];
                if (val < mv) {
                    heap[wave][lane][mi] = val;
                    mv = -BIGV;
                    #pragma unroll 1
                    for (int tt = 0; tt < TOPK; ++tt) {
                        const float h = heap[wave][lane][tt];
                        if (h > mv) { mv = h; mi = tt; }
                    }
                }
            }
            maxVal = mv;
            maxIdx = mi;
        }
        __syncthreads();
    }

    // Merge: global top-20 per row from the union of 8 per-wave top-20 sets.
    if (tid < 16) {
        float rsum = 0.0f;
        for (int sel = 0; sel < TOPK; ++sel) {
            float best = BIGV;
            int bw = 0, bt = 0;
            for (int w = 0; w < WAVES; ++w)
                for (int tt = 0; tt < TOPK; ++tt) {
                    const float h = heap[w][tid][tt];
                    if (h < best) { best = h; bw = w; bt = tt; }
                }
            if (best < 0.5f * BIGV) {
                rsum += best;
                heap[bw][tid][bt] = BIGV;
            }
        }
        red[tid] = rsum;
    }
    __syncthreads();
    if (tid == 0) {
        float s = 0.0f;
        for (int r = 0; r < 16; ++r) s += red[r];
        lp_part[blockIdx.x] = s;
    }
}

// ---------------------------------------------------------------- final combine
__global__ void final_kernel(const float* __restrict__ lp_part,
                             const float* __restrict__ ce_part,
                             float* __restrict__ out) {
    if (threadIdx.x == 0 && blockIdx.x == 0) {
        float lp = 0.0f;
        for (int i = 0; i < BATCH / 16; ++i) lp += lp_part[i];
        float ce = 0.0f;
        for (int i = 0; i < BATCH / 256; ++i) ce += ce_part[i];
        out[0] = 0.003f * 0.5f * lp + ce / (float)BATCH;
    }
}

// ---------------------------------------------------------------- launch
extern "C" void kernel_launch(void* const* d_in, const int* in_sizes, int n_in,
                              void* d_out, int out_size, void* d_ws, size_t ws_size,
                              hipStream_t stream) {
    const float* x_soft = (const float*)d_in[0];
    const float* x_feat = (const float*)d_in[1];
    const int*   yv     = (const int*)d_in[2];

    float* ws      = (float*)d_ws;
    float* sq      = ws;                       // BATCH floats
    float* lp_part = ws + BATCH;               // BATCH/16 = 512 floats
    float* ce_part = ws + BATCH + BATCH / 16;  // BATCH/256 = 32 floats

    norms_kernel<<<BATCH, 256, 0, stream>>>(x_feat, sq);
    ce_kernel<<<BATCH / 256, 256, 0, stream>>>(x_soft, yv, ce_part);
    lp_kernel<<<BATCH / 16, 256, 0, stream>>>(x_feat, yv, sq, lp_part);
    final_kernel<<<1, 64, 0, stream>>>(lp_part, ce_part, (float*)d_out);
}